// IntraClassLoss_38139309588962
// MI455X (gfx1250) — compile-verified
//
#include <hip/hip_runtime.h>
#include <hip/hip_bf16.h>

// ---------------------------------------------------------------------------
// IntraClassLoss for MI455X (gfx1250, wave32).
// Memory-bound: 2 passes over 134 MB of embeddings (~268 MB @ 23.3 TB/s ~12us
// floor). fp32 WMMA (V_WMMA_F32_16X16X4_F32) computes 16x16 similarity tiles;
// only the diagonal is used, but the redundant MACs are free under the
// bandwidth roofline and keep full fp32 fidelity vs the reference.
// ---------------------------------------------------------------------------

#define D_DIM 128
#define C_DIM 1024
#define NORM_EPS 1e-12f

typedef __attribute__((ext_vector_type(2))) float v2f;
typedef __attribute__((ext_vector_type(8))) float v8f;

// ---- kernel 0: zero the accumulators (sums [C*D], counts [C], loss [C]) ----
__global__ void icl_init(float* __restrict__ sums, float* __restrict__ counts,
                         float* __restrict__ loss, int cd, int c) {
    int i = blockIdx.x * blockDim.x + threadIdx.x;
    if (i < cd) sums[i] = 0.0f;
    if (i < c) { counts[i] = 0.0f; loss[i] = 0.0f; }
}

// ---- kernel 1: normalize rows, scatter-add into per-class sums -------------
// One wave32 per row: each lane owns a float4 (32*4 = 128 dims).
__global__ void icl_pass1(const float* __restrict__ emb,
                          const int* __restrict__ labels,
                          float* __restrict__ sums, float* __restrict__ counts,
                          float* __restrict__ invn, int n) {
    int row  = (blockIdx.x * blockDim.x + threadIdx.x) >> 5;
    int lane = threadIdx.x & 31;
    if (row >= n) return;
    const float4 v = ((const float4*)(emb + (size_t)row * D_DIM))[lane];
    float ss = v.x * v.x + v.y * v.y + v.z * v.z + v.w * v.w;
#pragma unroll
    for (int o = 16; o > 0; o >>= 1) ss += __shfl_xor(ss, o, 32);
    const float inv = 1.0f / fmaxf(sqrtf(ss), NORM_EPS);
    const int lab = labels[row];
    float* srow = sums + (size_t)lab * D_DIM + lane * 4;
    atomicAdd(srow + 0, v.x * inv);
    atomicAdd(srow + 1, v.y * inv);
    atomicAdd(srow + 2, v.z * inv);
    atomicAdd(srow + 3, v.w * inv);
    if (lane == 0) { atomicAdd(&counts[lab], 1.0f); invn[row] = inv; }
}

// ---- kernel 2: prototypes = l2norm(sums / max(count,1)) --------------------
// One wave32 per class.
__global__ void icl_protos(const float* __restrict__ sums,
                           const float* __restrict__ counts,
                           float* __restrict__ proto) {
    int c    = (blockIdx.x * blockDim.x + threadIdx.x) >> 5;
    int lane = threadIdx.x & 31;
    if (c >= C_DIM) return;
    const float rcnt = 1.0f / fmaxf(counts[c], 1.0f);
    float4 v = ((const float4*)(sums + (size_t)c * D_DIM))[lane];
    v.x *= rcnt; v.y *= rcnt; v.z *= rcnt; v.w *= rcnt;
    float ss = v.x * v.x + v.y * v.y + v.z * v.z + v.w * v.w;
#pragma unroll
    for (int o = 16; o > 0; o >>= 1) ss += __shfl_xor(ss, o, 32);
    const float inv = 1.0f / fmaxf(sqrtf(ss), NORM_EPS);
    float4 p; p.x = v.x * inv; p.y = v.y * inv; p.z = v.z * inv; p.w = v.w * inv;
    ((float4*)(proto + (size_t)c * D_DIM))[lane] = p;
}

// ---- kernel 3: WMMA similarity pass ----------------------------------------
// One wave32 per 16-sample tile. A = normalized embeddings (16x128),
// B = gathered prototypes^T (128x16). D = A*B accumulated via 32x
// V_WMMA_F32_16X16X4_F32; diagonal element i = sim of sample i.
//
// fp32 WMMA A-layout (16x4): lane j  (j<16) holds A[j][k], A[j][k+1];
//                            lane j+16     holds A[j][k+2], A[j][k+3].
// fp32 WMMA B-layout (4x16): lane j  (j<16) holds B[k][j],   B[k+1][j];
//                            lane j+16     holds B[k+2][j], B[k+3][j].
// => per lane: 2 consecutive floats at element d+2*hi of emb row j / proto row.
__global__ void icl_pass3(const float* __restrict__ emb,
                          const int* __restrict__ labels,
                          const float* __restrict__ proto,
                          const float* __restrict__ invn,
                          float* __restrict__ loss, int ntiles) {
    const int tile = (blockIdx.x * blockDim.x + threadIdx.x) >> 5;
    const int lane = threadIdx.x & 31;
    if (tile >= ntiles) return;              // whole wave takes same branch
    const int j  = lane & 15;
    const int hi = lane >> 4;
    const int s  = tile * 16 + j;
    const int lab = labels[s];
    const float inv = invn[s];
    const float* erow = emb   + (size_t)s   * D_DIM + 2 * hi;
    const float* prow = proto + (size_t)lab * D_DIM + 2 * hi;
    __builtin_prefetch(prow, 0, 0);          // global_prefetch_b8

    v8f acc = {};
#pragma unroll
    for (int d = 0; d < D_DIM; d += 4) {
        v2f a, b;
        a.x = erow[d] * inv; a.y = erow[d + 1] * inv;   // normalized emb
        b.x = prow[d];       b.y = prow[d + 1];         // prototype
        acc = __builtin_amdgcn_wmma_f32_16x16x4_f32(
            false, a, false, b, (short)0, acc, false, false);
    }

    // C/D layout: acc[r] @ lane l<16 -> (M=r, N=l); lane l>=16 -> (M=r+8, N=l-16).
    // Diagonal lives in lanes 0..7 (r=lane) and lanes 24..31 (r=lane-24).
    const int r = lane & 7;
    float sim = acc[0];
    sim = (r == 1) ? acc[1] : sim;
    sim = (r == 2) ? acc[2] : sim;
    sim = (r == 3) ? acc[3] : sim;
    sim = (r == 4) ? acc[4] : sim;
    sim = (r == 5) ? acc[5] : sim;
    sim = (r == 6) ? acc[6] : sim;
    sim = (r == 7) ? acc[7] : sim;
    const bool active = (lane < 8) || (lane >= 24);   // each sample exactly once
    if (active) atomicAdd(&loss[lab], 1.0f - sim);
}

// ---- kernel 4: final reduction over classes --------------------------------
__global__ void icl_final(const float* __restrict__ loss,
                          const float* __restrict__ counts,
                          float* __restrict__ out) {
    __shared__ float s_num[C_DIM];
    __shared__ float s_den[C_DIM];
    const int c = threadIdx.x;
    const float cnt   = counts[c];
    const float valid = (cnt >= 2.0f) ? 1.0f : 0.0f;
    const float per   = loss[c] / fmaxf(cnt, 1.0f);
    s_num[c] = per * valid;
    s_den[c] = valid;
    __syncthreads();
    for (int stride = C_DIM / 2; stride > 0; stride >>= 1) {
        if (c < stride) {
            s_num[c] += s_num[c + stride];
            s_den[c] += s_den[c + stride];
        }
        __syncthreads();
    }
    if (c == 0) {
        const float nv = s_den[0];
        out[0] = (nv > 0.0f) ? (s_num[0] / fmaxf(nv, 1.0f)) : 0.0f;
    }
}

extern "C" void kernel_launch(void* const* d_in, const int* in_sizes, int n_in,
                              void* d_out, int out_size, void* d_ws, size_t ws_size,
                              hipStream_t stream) {
    const float* emb    = (const float*)d_in[0];
    const int*   labels = (const int*)d_in[1];
    const int N = in_sizes[0] / D_DIM;          // 262144

    // workspace layout (floats): sums[C*D] | proto[C*D] | counts[C] | loss[C] | invn[N]
    float* ws     = (float*)d_ws;
    float* sums   = ws;
    float* proto  = sums  + (size_t)C_DIM * D_DIM;
    float* counts = proto + (size_t)C_DIM * D_DIM;
    float* loss   = counts + C_DIM;
    float* invn   = loss   + C_DIM;
    float* out    = (float*)d_out;

    const int CD = C_DIM * D_DIM;
    icl_init<<<(CD + 255) / 256, 256, 0, stream>>>(sums, counts, loss, CD, C_DIM);

    // 8 waves per 256-thread block, one wave per row
    icl_pass1<<<(N + 7) / 8, 256, 0, stream>>>(emb, labels, sums, counts, invn, N);

    icl_protos<<<(C_DIM + 7) / 8, 256, 0, stream>>>(sums, counts, proto);

    const int ntiles = N / 16;                  // N is a multiple of 16
    icl_pass3<<<(ntiles + 7) / 8, 256, 0, stream>>>(emb, labels, proto, invn, loss, ntiles);

    icl_final<<<1, C_DIM, 0, stream>>>(loss, counts, out);
}